// SDN_81054622810656
// MI455X (gfx1250) — compile-verified
//
#include <hip/hip_runtime.h>

// SDN room-acoustics recursion for MI455X (gfx1250, wave32).
// Single persistent workgroup; all delay-line state in LDS; 16-step time
// batching (valid because min propagation delay >> 16 samples); Householder
// scattering via v_wmma_f32_16x16x4_f32 on a zero-padded 32x32 block-diag.
// This revision makes the serial pp-chain and FIR-read phases branchless so
// the compiler emits v_cndmask/v_mul selects instead of per-step
// s_and_saveexec + s_wait_dscnt serialization.

#define N6     6
#define NM1    5
#define NLINES 30
#define LBUF   512
#define KP1    8
#define BSTEP  16
#define NBUF   43      // 1 src->mic + 6 src->node + 30 node->node + 6 node->mic
#define SMB    0
#define SNB    1
#define NNB    7
#define NMB    37

typedef float v2f __attribute__((ext_vector_type(2)));
typedef float v8f __attribute__((ext_vector_type(8)));

__launch_bounds__(128)
__global__ void sdn_kernel(const float* __restrict__ x,
                           const float* __restrict__ jf,    // (6,8)
                           const float* __restrict__ ju,    // (6,5)
                           const float* __restrict__ pw,    // (30,)
                           const float* __restrict__ perm,  // (30,30)
                           const float* __restrict__ d_sn,  // (6,)
                           const float* __restrict__ d_nn,  // (30,)
                           const float* __restrict__ d_nm,  // (6,)
                           const float* __restrict__ d_sm,  // (1,)
                           float* __restrict__ y, int T)
{
    const float G = 343.0f / 16000.0f;

    __shared__ float bufs[NBUF * LBUF];        // all 43 circular delay lines
    __shared__ float sbd[32][32];              // padded block-diag Householder
    __shared__ float pp_lds[32][BSTEP];        // incoming pressures, 16 steps
    __shared__ float pm_lds[32][BSTEP];        // scattered pressures (WMMA out)
    __shared__ float rd_lds[32][BSTEP];        // node->node FIR reads
    __shared__ float snr[N6][BSTEP];           // src->node reads
    __shared__ float xs[BSTEP];
    __shared__ float ypart[BSTEP][8];
    __shared__ float sgain[N6], micg[N6], fracsn[N6];
    __shared__ int   dintsn[N6];
    __shared__ float pwl[NLINES];
    __shared__ int   pidxl[NLINES];

    const int tid  = threadIdx.x;
    const int lane = tid & 31;
    const bool is_line = (lane < NLINES);
    const float lmask  = is_line ? 1.0f : 0.0f;   // branchless lane validity

    // ---- setup: zero buffers, derived constants ----
    for (int i = tid; i < NBUF * LBUF; i += 128) bufs[i] = 0.0f;

    for (int i = tid; i < 32 * 32; i += 128) {
        int r = i >> 5, c = i & 31;
        float v = 0.0f;
        if (r < NLINES && c < NLINES && (r / NM1 == c / NM1)) {
            int n = r / NM1, ri = r % NM1, ci = c % NM1;
            float nrm = 0.0f;
            for (int k = 0; k < NM1; ++k) nrm += ju[n * NM1 + k] * ju[n * NM1 + k];
            v = 2.0f * ju[n * NM1 + ri] * ju[n * NM1 + ci] / nrm - (ri == ci ? 1.0f : 0.0f);
        }
        sbd[r][c] = v;
    }
    if (tid < N6) {
        float dsn = d_sn[tid];
        sgain[tid] = 0.5f * G / dsn;
        micg[tid]  = 1.0f / (1.0f + d_nm[tid] / dsn);
        float del  = dsn / G;
        int   di   = (int)floorf(del);
        dintsn[tid] = di;
        fracsn[tid] = del - (float)di;
    }
    if (tid < NLINES) {
        pwl[tid] = pw[tid];
        int col = 0;
        for (int c = 0; c < NLINES; ++c)
            if (perm[tid * NLINES + c] > 0.5f) col = c;
        pidxl[tid] = col;
    }

    // per-thread constants: node->node line = lane (same mapping in every wave).
    // Invalid lanes (30,31) alias line 0 with all-zero taps -> contribute 0.
    const int lline = is_line ? lane : 0;
    const int lnode = lline / NM1;
    int dint_nn = 0; float frac_nn = 0.0f; float taps[KP1];
    {
        float del = d_nn[lline] / G;
        dint_nn = (int)floorf(del);
        frac_nn = del - (float)dint_nn;
        #pragma unroll
        for (int k = 0; k < KP1; ++k) taps[k] = lmask * jf[lnode * KP1 + k];
    }
    // output-phase constants: tid<112 -> (j = tid&15, q = tid>>4); q<6 node FIR, q==6 direct
    int dint_q = 0; float frac_q = 0.0f, gain_q = 0.0f; float tq[KP1];
    #pragma unroll
    for (int k = 0; k < KP1; ++k) tq[k] = 0.0f;
    if (tid < 112) {
        int q = tid >> 4;
        if (q < N6) {
            float del = d_nm[q] / G;
            dint_q = (int)floorf(del);
            frac_q = del - (float)dint_q;
            #pragma unroll
            for (int k = 0; k < KP1; ++k) tq[k] = jf[q * KP1 + k];
        } else {
            float dsm = d_sm[0];
            gain_q = 1.0f / dsm;
            float del = dsm / G;
            dint_q = (int)floorf(del);
            frac_q = del - (float)dint_q;
            tq[KP1 - 1] = 1.0f;  // single nominal-delay tap
        }
    }
    __syncthreads();

    // ---- preload WMMA A fragments (16x4 f32 layout: lanes 0-15 M, VGPR pair holds K/K+1) ----
    v2f afrag[2][8];
    {
        int m  = lane & 15;
        int kb = (lane >> 4) << 1;
        #pragma unroll
        for (int rb = 0; rb < 2; ++rb)
            #pragma unroll
            for (int c = 0; c < 8; ++c) {
                afrag[rb][c].x = sbd[rb * 16 + m][4 * c + kb];
                afrag[rb][c].y = sbd[rb * 16 + m][4 * c + kb + 1];
            }
    }
    const int myperm = is_line ? pidxl[lane] : 0;
    float carry = 0.0f;   // pp_next carried across blocks (wave 0 lanes)

    const int nblocks = (T + BSTEP - 1) / BSTEP;
    for (int b = 0; b < nblocks; ++b) {
        const int t0 = b * BSTEP;
        if (tid < BSTEP) xs[tid] = (t0 + tid < T) ? x[t0 + tid] : 0.0f;
        __syncthreads();

        // source injections for the whole block (min delay >> 16 -> safe)
        for (int idx = tid; idx < 112; idx += 128) {
            int j = idx & 15, q = idx >> 4;
            int w = (t0 + j) & (LBUF - 1);
            if (q < N6) bufs[(SNB + q) * LBUF + w] = sgain[q] * xs[j];
            else        bufs[SMB * LBUF + w] = xs[j];
        }
        __syncthreads();

        // node->node 8-tap fractional FIR reads: 30 lines x 16 steps, 4 waves over j.
        // Branchless: lanes 30,31 read line 0 with zero taps -> acc == 0.
        {
            int jw = tid >> 5;
            int base = (NNB + lline) * LBUF;
            #pragma unroll
            for (int jj = 0; jj < 4; ++jj) {
                int j = jw * 4 + jj;
                int tt = t0 + j - dint_nn;
                float acc = 0.0f;
                #pragma unroll
                for (int k = 0; k < KP1; ++k) {
                    int i0 = (tt - (KP1 - 1 - k)) & (LBUF - 1);
                    float v0 = bufs[base + i0];
                    float v1 = bufs[base + ((i0 - 1) & (LBUF - 1))];
                    acc += taps[k] * ((1.0f - frac_nn) * v0 + frac_nn * v1);
                }
                rd_lds[lane][j] = acc;
            }
        }
        // src->node single-tap interpolated reads
        for (int idx = tid; idx < N6 * BSTEP; idx += 128) {
            int n = idx >> 4, j = idx & 15;
            int base = (SNB + n) * LBUF;
            int i0 = (t0 + j - dintsn[n]) & (LBUF - 1);
            snr[n][j] = (1.0f - fracsn[n]) * bufs[base + i0]
                      + fracsn[n] * bufs[base + ((i0 - 1) & (LBUF - 1))];
        }
        __syncthreads();

        // wave 0: chain pp across the 16 steps (branchless), then WMMA scattering
        if (tid < 32) {
            float cprev = carry;
            #pragma unroll
            for (int j = 0; j < BSTEP; ++j) {
                float nxt = rd_lds[myperm][j];          // permuted next pp
                float ppv = lmask * (cprev + snr[lnode][j]);
                pp_lds[lane][j] = ppv;                  // rows 30,31 stay 0
                cprev = nxt;
            }
            carry = cprev;

            // PM(32x16) = Sbd(32x32) @ PP(32x16): 2 row tiles x 8 K-chunks of 4
            int col  = lane & 15;
            int rsel = (lane >> 4) << 1;
            #pragma unroll
            for (int rb = 0; rb < 2; ++rb) {
                v8f acc = {};
                #pragma unroll
                for (int c = 0; c < 8; ++c) {
                    v2f bfrag;
                    int r0 = 4 * c + rsel;
                    bfrag.x = pp_lds[r0][col];
                    bfrag.y = pp_lds[r0 + 1][col];
                    acc = __builtin_amdgcn_wmma_f32_16x16x4_f32(
                        /*neg_a=*/false, afrag[rb][c],
                        /*neg_b=*/false, bfrag,
                        /*c_mod=*/(short)0, acc,
                        /*reuse_a=*/false, /*reuse_b=*/false);
                }
                int mbase = rb * 16 + ((lane >> 4) << 3);
                #pragma unroll
                for (int v = 0; v < 8; ++v) pm_lds[mbase + v][col] = acc[v];
            }
        }
        __syncthreads();

        // write scattered pressures into node->node lines (store must stay predicated)
        {
            int jw = tid >> 5;
            #pragma unroll
            for (int jj = 0; jj < 4; ++jj) {
                int j = jw * 4 + jj;
                if (is_line)
                    bufs[(NNB + lane) * LBUF + ((t0 + j) & (LBUF - 1))] = pm_lds[lane][j];
            }
        }
        // node pressure -> node->mic lines
        for (int idx = tid; idx < N6 * BSTEP; idx += 128) {
            int n = idx >> 4, j = idx & 15;
            float s = 0.0f;
            #pragma unroll
            for (int i = 0; i < NM1; ++i) s += pwl[n * NM1 + i] * pm_lds[n * NM1 + i][j];
            bufs[(NMB + n) * LBUF + ((t0 + j) & (LBUF - 1))] = micg[n] * s;
        }
        __syncthreads();

        // mic output: 6 node FIRs + direct path, 112 parallel partials
        if (tid < 112) {
            int j = tid & 15, q = tid >> 4;
            int base = (q < N6 ? (NMB + q) : SMB) * LBUF;
            int tt = t0 + j - dint_q;
            float acc = 0.0f;
            #pragma unroll
            for (int k = 0; k < KP1; ++k) {
                int i0 = (tt - (KP1 - 1 - k)) & (LBUF - 1);
                float v0 = bufs[base + i0];
                float v1 = bufs[base + ((i0 - 1) & (LBUF - 1))];
                acc += tq[k] * ((1.0f - frac_q) * v0 + frac_q * v1);
            }
            ypart[j][q] = (q == N6) ? gain_q * acc : acc;
        }
        __syncthreads();
        if (tid < BSTEP && t0 + tid < T) {
            float s = 0.0f;
            #pragma unroll
            for (int q = 0; q < 7; ++q) s += ypart[tid][q];
            y[t0 + tid] = s;
        }
        __syncthreads();
    }
}

extern "C" void kernel_launch(void* const* d_in, const int* in_sizes, int n_in,
                              void* d_out, int out_size, void* d_ws, size_t ws_size,
                              hipStream_t stream) {
    (void)n_in; (void)d_ws; (void)ws_size;
    const float* x    = (const float*)d_in[0];
    const float* jf   = (const float*)d_in[1];
    const float* ju   = (const float*)d_in[2];
    const float* pw   = (const float*)d_in[3];
    const float* perm = (const float*)d_in[4];
    const float* dsn  = (const float*)d_in[5];
    const float* dnn  = (const float*)d_in[6];
    const float* dnm  = (const float*)d_in[7];
    const float* dsm  = (const float*)d_in[8];
    float* y = (float*)d_out;
    int T = in_sizes[0];
    (void)out_size;
    sdn_kernel<<<1, 128, 0, stream>>>(x, jf, ju, pw, perm, dsn, dnn, dnm, dsm, y, T);
}